// HierarchicalClusteringLayer_10634339025516
// MI455X (gfx1250) — compile-verified
//
#include <hip/hip_runtime.h>
#include <hip/hip_bf16.h>
#include <math.h>

// ---------------------------------------------------------------------------
//   x: [16384,1024] f32; W1:[1024,256]; b1:[256]; W2:[256,32]; b2:[32]
//   h1 = gelu(gelu(x)@W1+b1) ; feats = h1@W2+b2 ; per-batch Ward -> int32
// ---------------------------------------------------------------------------

typedef __attribute__((ext_vector_type(2))) float v2f;
typedef __attribute__((ext_vector_type(4))) float v4f;
typedef __attribute__((ext_vector_type(8))) float v8f;

__device__ __forceinline__ float gelu_exact(float v) {
    return 0.5f * v * (1.0f + erff(v * 0.70710678118654752440f));
}

__device__ __forceinline__ void async_b128(uint32_t lds_off, const void* gptr) {
    // GLOBAL_LOAD_ASYNC_TO_LDS_B128 (GV mode): copies 16 B from vaddr into LDS at vdst, tracked by ASYNCcnt
    uint64_t ga = (uint64_t)(uintptr_t)gptr;
    asm volatile("global_load_async_to_lds_b128 %0, %1, off"
                 :: "v"(lds_off), "v"(ga) : "memory");
}
__device__ __forceinline__ void wait_async0() {
    asm volatile("s_wait_asynccnt 0x0" ::: "memory");
}

// ---------------------------------------------------------------------------
// Kernel 0: one-time weight swizzle into fragment-pair order.
// w1p flat idx = (((ny*32+c)*16+p)*128+col)*2+sel, value = W1[c*32+2p+sel][ny*128+col]
// w2p flat idx = (p*32+col)*2+sel,                 value = W2[2p+sel][col]
// ---------------------------------------------------------------------------
__global__ __launch_bounds__(256) void prep_kernel(
    const float* __restrict__ W1, const float* __restrict__ W2,
    float* __restrict__ w1p, float* __restrict__ w2p)
{
    int g = blockIdx.x * 256 + threadIdx.x;
    if (g < 131072) {
        int col = g & 127, p = (g >> 7) & 15, c = (g >> 11) & 31, ny = g >> 16;
        int row = c * 32 + 2 * p;
        float a = W1[(size_t)row * 256 + ny * 128 + col];
        float b = W1[(size_t)(row + 1) * 256 + ny * 128 + col];
        w1p[(size_t)g * 2]     = a;
        w1p[(size_t)g * 2 + 1] = b;
    } else if (g < 131072 + 4096) {
        int q = g - 131072;
        int col = q & 31, p = q >> 5;
        w2p[(size_t)q * 2]     = W2[(size_t)(2 * p) * 32 + col];
        w2p[(size_t)q * 2 + 1] = W2[(size_t)(2 * p + 1) * 32 + col];
    }
}

// ---------------------------------------------------------------------------
// Kernel 1: h1 = gelu(gelu(x) @ W1 + b1)
// grid (256,2), block 256 (8 waves). Block tile 64x128; wave: strip w>>1,
// col-half w&1, 4 accumulators. Double-buffered: async W1 chunk + register-
// staged gelu(x) chunk for c+1 overlap the 32 WMMAs of chunk c.
// ---------------------------------------------------------------------------
#define XS_S 36    // floats/row: 16B aligned, conflict-free A-frag reads
#define WP_S 144   // float2 per p-row: 2*144 mod 64 = 32 -> halves disjoint

__global__ __launch_bounds__(256) void mlp1_kernel(
    const float* __restrict__ x, const float* __restrict__ w1p,
    const float* __restrict__ b1, float* __restrict__ h1)
{
    __shared__ float xs[2][64 * XS_S];          // gelu(x) tiles  (2 x 9.2 KB)
    __shared__ float ws1p[2][16 * WP_S * 2];    // W1 pair tiles  (2 x 18 KB)

    const int tid  = threadIdx.x;
    const int lane = tid & 31;
    const int w    = tid >> 5;
    const int strip   = w >> 1;
    const int halfcol = w & 1;
    const int m0 = blockIdx.x * 64;
    const int ny = blockIdx.y;          // column half of the 256-wide output

    const uint32_t ws1p_off[2] = { (uint32_t)(uintptr_t)&ws1p[0][0],
                                   (uint32_t)(uintptr_t)&ws1p[1][0] };
    const uint32_t xs_off[2]   = { (uint32_t)(uintptr_t)&xs[0][0],
                                   (uint32_t)(uintptr_t)&xs[1][0] };
    (void)xs_off;

    v8f acc[4];
#pragma unroll
    for (int ct = 0; ct < 4; ++ct) acc[ct] = (v8f){0.f,0.f,0.f,0.f,0.f,0.f,0.f,0.f};

    const int lrow  = strip * 16 + (lane & 15);
    const int khalf = (lane >> 4) << 1;

    // x register-staging coordinates (2 float4 per thread per chunk)
    const int f0 = tid, f1 = tid + 256;
    const size_t xo0 = (size_t)(m0 + (f0 >> 3)) * 1024 + (f0 & 7) * 4;
    const size_t xo1 = (size_t)(m0 + (f1 >> 3)) * 1024 + (f1 & 7) * 4;
    const int xd0 = (f0 >> 3) * XS_S + (f0 & 7) * 4;
    const int xd1 = (f1 >> 3) * XS_S + (f1 & 7) * 4;

    // --- staging helpers (inlined by hand via lambdas) ---
    auto issue_w1 = [&](int c, int buf) {
        const float* gbase = w1p + (size_t)((ny * 32 + c) * 16) * 256;
#pragma unroll
        for (int i = 0; i < 4; ++i) {
            int f = tid + i * 256;              // 1024 B128 transfers / chunk
            int p = f >> 6, seg = f & 63;
            async_b128(ws1p_off[buf] + (uint32_t)((p * WP_S + seg * 2) * 8),
                       gbase + (size_t)(p * 256 + seg * 4));
        }
    };
    auto stage_x = [&](int c, int buf) {
        const int kc = c * 32;
        v4f t0 = *(const v4f*)(x + xo0 + kc);
        v4f t1 = *(const v4f*)(x + xo1 + kc);
        v4f g0, g1;
#pragma unroll
        for (int q = 0; q < 4; ++q) { g0[q] = gelu_exact(t0[q]); g1[q] = gelu_exact(t1[q]); }
        *(v4f*)&xs[buf][xd0] = g0;
        *(v4f*)&xs[buf][xd1] = g1;
    };

    // prologue: chunk 0 -> buf 0
    issue_w1(0, 0);
    stage_x(0, 0);
    wait_async0();
    __syncthreads();

    for (int c = 0; c < 32; ++c) {
        const int buf = c & 1;
        if (c < 31) {                      // overlap next chunk with compute
            issue_w1(c + 1, buf ^ 1);
            stage_x(c + 1, buf ^ 1);
        }
#pragma unroll
        for (int kk = 0; kk < 32; kk += 4) {
            const int kb = kk + khalf;
            v2f a;
            a.x = xs[buf][lrow * XS_S + kb];
            a.y = xs[buf][lrow * XS_S + kb + 1];
            const int p = (kk >> 1) + (lane >> 4);
#pragma unroll
            for (int ct = 0; ct < 4; ++ct) {
                const int nb = halfcol * 64 + ct * 16 + (lane & 15);
                v2f bf = *(const v2f*)&ws1p[buf][(p * WP_S + nb) * 2];
                acc[ct] = __builtin_amdgcn_wmma_f32_16x16x4_f32(
                    false, a, false, bf, (short)0, acc[ct], false, false);
            }
        }
        wait_async0();
        __syncthreads();
    }

    // epilogue: +b1, gelu, store
#pragma unroll
    for (int ct = 0; ct < 4; ++ct) {
        const int col = ny * 128 + halfcol * 64 + ct * 16 + (lane & 15);
        const float bias = b1[col];
#pragma unroll
        for (int r = 0; r < 8; ++r) {
            int row = m0 + strip * 16 + r + ((lane >> 4) << 3);
            h1[(size_t)row * 256 + col] = gelu_exact(acc[ct][r] + bias);
        }
    }
}

// ---------------------------------------------------------------------------
// Kernel 2: feats = h1 @ W2 + b2. grid 1024, block 64 (2 waves, one N-tile
// each). h1 strip + pair-swizzled W2 image async-copied to LDS.
// ---------------------------------------------------------------------------
#define H1S_S 260  // 16B-aligned rows; bank phases of lane halves disjoint

__global__ __launch_bounds__(64) void mlp2_kernel(
    const float* __restrict__ h1, const float* __restrict__ w2p,
    const float* __restrict__ b2, float* __restrict__ feats)
{
    __shared__ float h1s[16 * H1S_S];   // 16.6 KB
    __shared__ float w2s[8192];         // 32 KB, exact image of w2p

    const int tid  = threadIdx.x;
    const int lane = tid & 31;
    const int w    = tid >> 5;
    const int m0   = blockIdx.x * 16;

    const uint32_t h1s_off = (uint32_t)(uintptr_t)&h1s[0];
    const uint32_t w2s_off = (uint32_t)(uintptr_t)&w2s[0];

    // async-stage h1 strip: 1024 B128 / 64 threads = 16 each
#pragma unroll
    for (int i = 0; i < 16; ++i) {
        int f = tid + i * 64;
        int row = f >> 6, seg = f & 63;
        async_b128(h1s_off + (uint32_t)((row * H1S_S + seg * 4) * 4),
                   h1 + (size_t)(m0 + row) * 256 + seg * 4);
    }
    // async-stage swizzled W2: 2048 B128 / 64 threads = 32 each, contiguous
#pragma unroll
    for (int i = 0; i < 32; ++i) {
        int f = tid + i * 64;
        async_b128(w2s_off + (uint32_t)(f * 16), w2p + (size_t)f * 4);
    }
    wait_async0();
    __syncthreads();

    v8f acc = (v8f){0.f,0.f,0.f,0.f,0.f,0.f,0.f,0.f};
    const int khalf = (lane >> 4) << 1;
    const int n = w * 16 + (lane & 15);

#pragma unroll 8
    for (int k4 = 0; k4 < 256; k4 += 4) {
        const int kb = k4 + khalf;
        v2f a;
        a.x = h1s[(lane & 15) * H1S_S + kb];
        a.y = h1s[(lane & 15) * H1S_S + kb + 1];
        const int p = (k4 >> 1) + (lane >> 4);
        v2f bf = *(const v2f*)&w2s[(p * 32 + n) * 2];
        acc = __builtin_amdgcn_wmma_f32_16x16x4_f32(
            false, a, false, bf, (short)0, acc, false, false);
    }

    const float bias = b2[n];
#pragma unroll
    for (int r = 0; r < 8; ++r) {
        int row = m0 + r + ((lane >> 4) << 3);
        feats[(size_t)row * 32 + n] = acc[r] + bias;
    }
}

// ---------------------------------------------------------------------------
// Kernel 3: greedy Ward clustering, one block per batch (128 pts, 32-d, K=8).
// Full cost matrix + centroids in dynamic LDS (~84 KB of the 320 KB/WGP).
// Incremental updates; argmin tie-break = smallest flat index (jnp.argmin).
// ---------------------------------------------------------------------------
__global__ __launch_bounds__(128) void ward_kernel(
    const float* __restrict__ feats, int* __restrict__ out)
{
    extern __shared__ float smem[];
    float* cents = smem;                 // [128][33]
    float* Dm    = cents + 128 * 33;     // [128][129]
    float* sizes = Dm + 128 * 129;       // [128]
    float* sqn   = sizes + 128;          // [128]
    float* redv  = sqn + 128;            // [128]
    int*   redi  = (int*)(redv + 128);   // [128]
    int*   rank  = redi + 128;           // [128]

    const int b = blockIdx.x;
    const int t = threadIdx.x;
    const float INF = __builtin_inff();

    float s2 = 0.f;
#pragma unroll 8
    for (int d = 0; d < 32; ++d) {
        float c = feats[((size_t)b * 128 + t) * 32 + d];
        cents[t * 33 + d] = c;
        s2 += c * c;
    }
    sqn[t] = s2;
    sizes[t] = 1.f;
    int lab = t;
    __syncthreads();

    {
        const float st = sizes[t];
        for (int j = 0; j < 128; ++j) {
            float v;
            if (j == t) v = INF;
            else {
                float dot = 0.f;
#pragma unroll 8
                for (int d = 0; d < 32; ++d)
                    dot += cents[t * 33 + d] * cents[j * 33 + d];
                float d2 = fmaxf(sqn[t] + sqn[j] - 2.f * dot, 0.f);
                float sj = sizes[j];
                v = (st * sj / (st + sj + 1e-30f)) * d2;
            }
            Dm[t * 129 + j] = v;
        }
    }
    __syncthreads();

    for (int iter = 0; iter < 120; ++iter) {
        float bv = INF; int bj = 0;
        for (int j = 0; j < 128; ++j) {
            float v = Dm[t * 129 + j];
            if (v < bv) { bv = v; bj = j; }
        }
        redv[t] = bv; redi[t] = t * 128 + bj;
        __syncthreads();
        for (int s = 64; s > 0; s >>= 1) {
            if (t < s) {
                float ov = redv[t + s]; int oi = redi[t + s];
                if (ov < redv[t] || (ov == redv[t] && oi < redi[t])) {
                    redv[t] = ov; redi[t] = oi;
                }
            }
            __syncthreads();
        }
        const int flat = redi[0];
        const int i = flat >> 7, j = flat & 127;
        const int i2 = min(i, j), j2 = max(i, j);
        const float si = sizes[i2], sj = sizes[j2];
        __syncthreads();

        if (t < 32) {
            cents[i2 * 33 + t] =
                (cents[i2 * 33 + t] * si + cents[j2 * 33 + t] * sj) / (si + sj);
        }
        if (t == 0) { sizes[i2] = si + sj; sizes[j2] = 0.f; }
        if (lab == j2) lab = i2;
        __syncthreads();

        if (t == 0) {
            float s = 0.f;
#pragma unroll 8
            for (int d = 0; d < 32; ++d) { float c = cents[i2 * 33 + d]; s += c * c; }
            sqn[i2] = s;
        }
        __syncthreads();

        {
            const float snew = sizes[i2];
            const float stt  = sizes[t];
            float v;
            if (t == i2 || stt <= 0.f) v = INF;
            else {
                float dot = 0.f;
#pragma unroll 8
                for (int d = 0; d < 32; ++d)
                    dot += cents[i2 * 33 + d] * cents[t * 33 + d];
                float d2 = fmaxf(sqn[i2] + sqn[t] - 2.f * dot, 0.f);
                v = (snew * stt / (snew + stt + 1e-30f)) * d2;
            }
            Dm[i2 * 129 + t] = v;
            Dm[t * 129 + i2] = v;
            Dm[j2 * 129 + t] = INF;
            Dm[t * 129 + j2] = INF;
        }
        __syncthreads();
    }

    if (t == 0) {
        int c = 0;
        for (int idx = 0; idx < 128; ++idx) {
            if (sizes[idx] > 0.f) ++c;
            rank[idx] = c - 1;
        }
    }
    __syncthreads();
    out[(size_t)b * 128 + t] = rank[lab];
}

// ---------------------------------------------------------------------------
extern "C" void kernel_launch(void* const* d_in, const int* in_sizes, int n_in,
                              void* d_out, int out_size, void* d_ws, size_t ws_size,
                              hipStream_t stream) {
    (void)in_sizes; (void)n_in; (void)out_size; (void)ws_size;
    const float* x  = (const float*)d_in[0];
    const float* W1 = (const float*)d_in[1];
    const float* b1 = (const float*)d_in[2];
    const float* W2 = (const float*)d_in[3];
    const float* b2 = (const float*)d_in[4];
    int* out = (int*)d_out;

    float* h1    = (float*)d_ws;                    // [16384,256]
    float* feats = h1    + (size_t)16384 * 256;     // [16384,32]
    float* w1p   = feats + (size_t)16384 * 32;      // 262144 floats (swizzled W1)
    float* w2p   = w1p   + (size_t)262144;          // 8192 floats  (swizzled W2)

    prep_kernel<<<528, 256, 0, stream>>>(W1, W2, w1p, w2p);
    mlp1_kernel<<<dim3(256, 2), 256, 0, stream>>>(x, w1p, b1, h1);
    mlp2_kernel<<<1024, 64, 0, stream>>>(h1, w2p, b2, feats);

    const size_t smem_bytes =
        (size_t)(128 * 33 + 128 * 129 + 128 * 3) * sizeof(float) +
        (size_t)(128 * 2) * sizeof(int);            // ~84 KB dynamic LDS
    ward_kernel<<<128, 128, smem_bytes, stream>>>(feats, out);
}